// EncoderLEConv_80015240725026
// MI455X (gfx1250) — compile-verified
//
#include <hip/hip_runtime.h>
#include <hip/hip_bf16.h>

#define N_NODES 50000
#define N_EDGES 800000
#define D 64

typedef float v2f __attribute__((ext_vector_type(2)));
typedef float v8f __attribute__((ext_vector_type(8)));

__device__ __forceinline__ v8f wmma4(v2f a, v2f b, v8f c) {
    // D = A(16x4, f32) * B(4x16, f32) + C(16x16, f32)
    return __builtin_amdgcn_wmma_f32_16x16x4_f32(
        /*neg_a=*/false, a, /*neg_b=*/false, b,
        /*c_mod=*/(short)0, c, /*reuse_a=*/false, /*reuse_b=*/false);
}

// ---------------------------------------------------------------------------
// One-time per-launch preprocessing
// ---------------------------------------------------------------------------

// Pack edge_index (2 x int64, 6.4MB apart) into int2{src,dst} (one 8B load).
__global__ __launch_bounds__(256)
void edge_pack_kernel(const long long* __restrict__ ei, int2* __restrict__ ep, int nE) {
    int e = blockIdx.x * blockDim.x + threadIdx.x;
    if (e < nE) {
        ep[e] = make_int2((int)ei[e], (int)ei[(size_t)nE + e]);
    }
}

__global__ __launch_bounds__(256)
void deg_zero_kernel(float* __restrict__ deg, int n) {
    int i = blockIdx.x * blockDim.x + threadIdx.x;
    if (i < n) deg[i] = 0.0f;
}

__global__ __launch_bounds__(256)
void deg_count_kernel(const int2* __restrict__ ep, float* __restrict__ deg, int nE) {
    int e = blockIdx.x * blockDim.x + threadIdx.x;
    if (e < nE) atomicAdd(&deg[ep[e].y], 1.0f);
}

// Repack a 64x64 row-major weight into WMMA-B-fragment order:
// entry t = (ktile*2 + h)*64 + c  ->  float2{ W[(4*ktile+2h)*64+c], W[(4*ktile+2h+1)*64+c] }
// so a lane's B fragment for k-step k0 is one contiguous b64 load.
__global__ __launch_bounds__(256)
void w_repack_kernel(const float* __restrict__ W, float* __restrict__ Wp) {
    int t = blockIdx.x * blockDim.x + threadIdx.x;   // 2048 float2 entries
    if (t < 2048) {
        int c     = t & 63;
        int kh    = t >> 6;        // ktile*2 + h
        int h     = kh & 1;
        int ktile = kh >> 1;
        int k     = ktile * 4 + 2 * h;
        ((float2*)Wp)[t] = make_float2(W[(size_t)k * D + c], W[(size_t)(k + 1) * D + c]);
    }
}

// ---------------------------------------------------------------------------
// Fused GEMM: a = x@W1 + b1 ; agg = x@W3 + b3 - deg * (x@W2)
// Block = 256 threads = 8 waves; block covers 32 rows x 64 cols.
// Wave w: mtile = w>>2 (0..1), ntile = w&3 (0..3) -> one 16x16 tile each.
// Weights are in repacked (Wp) layout.
// ---------------------------------------------------------------------------
__global__ __launch_bounds__(256)
void leconv_gemm3_kernel(const float* __restrict__ X,
                         const float* __restrict__ Wp1, const float* __restrict__ b1,
                         const float* __restrict__ Wp2,
                         const float* __restrict__ Wp3, const float* __restrict__ b3,
                         const float* __restrict__ deg,
                         float* __restrict__ Aout, float* __restrict__ Agg)
{
    const int lane = threadIdx.x & 31;
    const int w    = threadIdx.x >> 5;
    const int m0   = blockIdx.x * 32 + (w >> 2) * 16;
    const int n0   = (w & 3) * 16;
    const int half = lane >> 4;      // 0 or 1
    const int l16  = lane & 15;

    int arow = m0 + l16;
    if (arow >= N_NODES) arow = N_NODES - 1;
    const float* xrow = X + (size_t)arow * D + 2 * half;

    const int bcol = n0 + l16;
    const int bbase = (half * 64 + bcol) * 2;   // float offset inside a k-tile slab (256 floats)

    v8f acc1 = {}; v8f acc2 = {}; v8f acc3 = {};

#pragma unroll
    for (int k0 = 0; k0 < D; k0 += 4) {
        v2f af  = *(const v2f*)(xrow + k0);
        v2f bf1 = *(const v2f*)(Wp1 + k0 * 64 + bbase);
        v2f bf2 = *(const v2f*)(Wp2 + k0 * 64 + bbase);
        v2f bf3 = *(const v2f*)(Wp3 + k0 * 64 + bbase);
        acc1 = wmma4(af, bf1, acc1);
        acc2 = wmma4(af, bf2, acc2);
        acc3 = wmma4(af, bf3, acc3);
    }

    const float bias1 = b1[bcol];
    const float bias3 = b3[bcol];

#pragma unroll
    for (int g = 0; g < 8; ++g) {
        int row = m0 + g + 8 * half;
        if (row < N_NODES) {
            float dg = deg[row];
            size_t off = (size_t)row * D + bcol;
            Aout[off] = acc1[g] + bias1;
            Agg[off]  = acc3[g] + bias3 - dg * acc2[g];
        }
    }
}

// ---------------------------------------------------------------------------
// Edge scatter: agg[dst] += a[src]
// 16 threads per edge, float4 per thread. Lane 0/16 of each half-wave loads the
// packed indices once; broadcast via __shfl (LDS permute path).
// ---------------------------------------------------------------------------
__global__ __launch_bounds__(256)
void leconv_scatter_kernel(const int2* __restrict__ ep,
                           const float* __restrict__ a,
                           float* __restrict__ agg, int nE)
{
    int t    = blockIdx.x * blockDim.x + threadIdx.x;
    int e    = t >> 4;            // edge index (grid sized exactly: nE*16 threads)
    int lane = threadIdx.x & 31;
    int f    = (t & 15) * 4;      // feature base
    if (e >= nE) return;

    int s, d;
    {
        int2 p = make_int2(0, 0);
        if ((lane & 15) == 0) p = ep[e];
        s = __shfl(p.x, lane & 16);
        d = __shfl(p.y, lane & 16);
    }

    const float4 v = *(const float4*)(a + (size_t)s * D + f);
    float* gp = agg + (size_t)d * D + f;
    atomicAdd(gp + 0, v.x);
    atomicAdd(gp + 1, v.y);
    atomicAdd(gp + 2, v.z);
    atomicAdd(gp + 3, v.w);
}

// ---------------------------------------------------------------------------
// GEMM + bias + relu: out = relu(X @ W + b)   (W in repacked layout)
// ---------------------------------------------------------------------------
__global__ __launch_bounds__(256)
void leconv_gemm_relu_kernel(const float* __restrict__ X,
                             const float* __restrict__ Wp, const float* __restrict__ b,
                             float* __restrict__ Out)
{
    const int lane = threadIdx.x & 31;
    const int w    = threadIdx.x >> 5;
    const int m0   = blockIdx.x * 32 + (w >> 2) * 16;
    const int n0   = (w & 3) * 16;
    const int half = lane >> 4;
    const int l16  = lane & 15;

    int arow = m0 + l16;
    if (arow >= N_NODES) arow = N_NODES - 1;
    const float* xrow = X + (size_t)arow * D + 2 * half;

    const int bcol = n0 + l16;
    const int bbase = (half * 64 + bcol) * 2;

    v8f acc = {};

#pragma unroll
    for (int k0 = 0; k0 < D; k0 += 4) {
        v2f af = *(const v2f*)(xrow + k0);
        v2f bf = *(const v2f*)(Wp + k0 * 64 + bbase);
        acc = wmma4(af, bf, acc);
    }

    const float bias = b[bcol];

#pragma unroll
    for (int g = 0; g < 8; ++g) {
        int row = m0 + g + 8 * half;
        if (row < N_NODES) {
            Out[(size_t)row * D + bcol] = fmaxf(acc[g] + bias, 0.0f);
        }
    }
}

// ---------------------------------------------------------------------------
// Host-side orchestration
// ---------------------------------------------------------------------------
extern "C" void kernel_launch(void* const* d_in, const int* in_sizes, int n_in,
                              void* d_out, int out_size, void* d_ws, size_t ws_size,
                              hipStream_t stream)
{
    const float*     x  = (const float*)d_in[0];
    const long long* ei = (const long long*)d_in[1];

    // per-layer params: W1, b1, W2, W3, b3, Wl, bl  (indices 2..22)
    const float* P[21];
    for (int i = 0; i < 21; ++i) P[i] = (const float*)d_in[2 + i];

    float* out = (float*)d_out;

    // Workspace layout (floats):
    //   epack  : 800000 int2          = 1,600,000 i32
    //   Wp[12] : 12 * 4096 floats     =    49,152
    //   deg    : N                    =    50,000
    //   a, agg, h : N*D each          = 3 * 3,200,000
    float* ws    = (float*)d_ws;
    int2*  epack = (int2*)ws;
    float* wp    = ws + (size_t)2 * N_EDGES;
    float* deg   = wp + 12 * 4096;
    float* a     = deg + N_NODES;
    float* agg   = a + (size_t)N_NODES * D;
    float* h     = agg + (size_t)N_NODES * D;

    const dim3 blk(256);
    const int gemmGrid = (N_NODES + 31) / 32;
    const int scatGrid = (N_EDGES * 16 + 255) / 256;   // exact: 50000 blocks

    // --- one-time preprocessing (deterministic, re-done every launch) ---
    edge_pack_kernel<<<(N_EDGES + 255) / 256, blk, 0, stream>>>(ei, epack, N_EDGES);
    deg_zero_kernel<<<(N_NODES + 255) / 256, blk, 0, stream>>>(deg, N_NODES);
    deg_count_kernel<<<(N_EDGES + 255) / 256, blk, 0, stream>>>(epack, deg, N_EDGES);

    // Repack the 4 GEMM weights of each layer (W1, W2, W3, Wl)
    float* wpp[12];
    {
        const int widx[4] = {0, 2, 3, 5};   // offsets of W1, W2, W3, Wl within a layer's params
        for (int l = 0; l < 3; ++l) {
            for (int j = 0; j < 4; ++j) {
                int slot = l * 4 + j;
                wpp[slot] = wp + (size_t)slot * 4096;
                w_repack_kernel<<<8, blk, 0, stream>>>(P[l * 7 + widx[j]], wpp[slot]);
            }
        }
    }

    // --- 3 LEConv layers ---
    const float* cur = x;
    for (int l = 0; l < 3; ++l) {
        const float* b1 = P[l * 7 + 1];
        const float* b3 = P[l * 7 + 4];
        const float* bl = P[l * 7 + 6];
        const float* Wp1 = wpp[l * 4 + 0];
        const float* Wp2 = wpp[l * 4 + 1];
        const float* Wp3 = wpp[l * 4 + 2];
        const float* Wpl = wpp[l * 4 + 3];

        leconv_gemm3_kernel<<<gemmGrid, blk, 0, stream>>>(cur, Wp1, b1, Wp2, Wp3, b3,
                                                          deg, a, agg);
        leconv_scatter_kernel<<<scatGrid, blk, 0, stream>>>(epack, a, agg, N_EDGES);

        float* dst = (l == 2) ? out : h;
        leconv_gemm_relu_kernel<<<gemmGrid, blk, 0, stream>>>(agg, Wpl, bl, dst);
        cur = h;
    }
}